// VisionMamba_88493506167058
// MI455X (gfx1250) — compile-verified
//
#include <hip/hip_runtime.h>
#include <math.h>

// ---------------- model dims ----------------
#define D_MODEL   384
#define D_INNER   768
#define D_STATE   16
#define DT_RANK   24
#define NXPROJ    56      // DT_RANK + 2*D_STATE
#define NXPAD     64      // padded x_proj N so every GEMM N is a multiple of 64
#define DEPTH     24
#define L_TOK     196
#define BATCH     4
#define M_ROWS    (BATCH * L_TOK)   // 784, multiple of 16
#define EPSF      1e-5f

typedef __attribute__((ext_vector_type(2))) float v2f;
typedef __attribute__((ext_vector_type(8))) float v8f;

__device__ __forceinline__ int rot_l(int l) {
  // 14x14 grid transpose; self-inverse (perm == inv)
  return (l % 14) * 14 + (l / 14);
}

// ---------------------------------------------------------------------------
// Guard-free fp32 WMMA GEMM:  out[m,n] = sum_k A[m,k] * W[n,k]   (X @ W^T)
// A: M x K row-major (lda), W: N x K row-major, out: M x N (ldo).
// Preconditions (arranged by the host): M % 16 == 0, N % 64 == 0, K % 8 == 0.
// One wave computes a 16(M) x 64(N) tile; k unrolled by 2 (8 WMMAs / iter).
// mode 0: plain store
// mode 1: v += aux0[n]; v = softplus(v)                (dt_proj)
// mode 2: store row scattered through grid transpose   (out_proj, rotate layers)
// mode 3: v += aux0[n] + aux1[(m%L)*D_MODEL + n]       (patch embed + pos)
// ---------------------------------------------------------------------------
__global__ void gemm_wmma_f32(const float* __restrict__ A, int lda,
                              const float* __restrict__ W,
                              float* __restrict__ out, int ldo,
                              int K, int mode,
                              const float* __restrict__ aux0,
                              const float* __restrict__ aux1) {
  const int lane = threadIdx.x & 31;
  const int m0   = blockIdx.x * 16;
  const int n0   = blockIdx.y * 64;
  const int kb   = (lane >> 4) * 2;   // lanes 0-15 -> K {0,1}; lanes 16-31 -> K {2,3}
  const int ncol = lane & 15;

  const float* __restrict__ ap  = A + (size_t)(m0 + (lane & 15)) * lda + kb;
  const float* __restrict__ wp0 = W + (size_t)(n0 + ncol) * K + kb;
  const float* __restrict__ wp1 = wp0 + (size_t)16 * K;
  const float* __restrict__ wp2 = wp0 + (size_t)32 * K;
  const float* __restrict__ wp3 = wp0 + (size_t)48 * K;

  v8f acc0 = {}, acc1 = {}, acc2 = {}, acc3 = {};

  for (int kk = 0; kk < K; kk += 8) {
    // issue all loads for this k-slab up front (10x global_load_b64)
    v2f a0  = *(const v2f*)(ap  + kk);
    v2f a1  = *(const v2f*)(ap  + kk + 4);
    v2f b00 = *(const v2f*)(wp0 + kk);
    v2f b01 = *(const v2f*)(wp0 + kk + 4);
    v2f b10 = *(const v2f*)(wp1 + kk);
    v2f b11 = *(const v2f*)(wp1 + kk + 4);
    v2f b20 = *(const v2f*)(wp2 + kk);
    v2f b21 = *(const v2f*)(wp2 + kk + 4);
    v2f b30 = *(const v2f*)(wp3 + kk);
    v2f b31 = *(const v2f*)(wp3 + kk + 4);
    if (kk + 32 < K) {
      __builtin_prefetch(ap + kk + 32, 0, 1);   // global_prefetch_b8
    }
    acc0 = __builtin_amdgcn_wmma_f32_16x16x4_f32(false, a0, false, b00, (short)0, acc0, false, false);
    acc1 = __builtin_amdgcn_wmma_f32_16x16x4_f32(false, a0, false, b10, (short)0, acc1, false, false);
    acc2 = __builtin_amdgcn_wmma_f32_16x16x4_f32(false, a0, false, b20, (short)0, acc2, false, false);
    acc3 = __builtin_amdgcn_wmma_f32_16x16x4_f32(false, a0, false, b30, (short)0, acc3, false, false);
    acc0 = __builtin_amdgcn_wmma_f32_16x16x4_f32(false, a1, false, b01, (short)0, acc0, false, false);
    acc1 = __builtin_amdgcn_wmma_f32_16x16x4_f32(false, a1, false, b11, (short)0, acc1, false, false);
    acc2 = __builtin_amdgcn_wmma_f32_16x16x4_f32(false, a1, false, b21, (short)0, acc2, false, false);
    acc3 = __builtin_amdgcn_wmma_f32_16x16x4_f32(false, a1, false, b31, (short)0, acc3, false, false);
  }

  // Epilogue. C/D layout: VGPR j -> (lanes 0-15: M=j, N=ncol) (lanes 16-31: M=8+j)
  const int mbase = m0 + (lane >> 4) * 8;
  v8f accs[4] = {acc0, acc1, acc2, acc3};
#pragma unroll
  for (int nt = 0; nt < 4; ++nt) {
    int n = n0 + nt * 16 + ncol;
#pragma unroll
    for (int j = 0; j < 8; ++j) {
      int m = mbase + j;
      float v = accs[nt][j];
      int orow = m;
      if (mode == 1) {
        v += aux0[n];
        v = (v > 20.0f) ? v : log1pf(__expf(v));
      } else if (mode == 2) {
        int b_ = m / L_TOK, l = m % L_TOK;
        orow = b_ * L_TOK + rot_l(l);
      } else if (mode == 3) {
        v += aux0[n] + aux1[(m % L_TOK) * D_MODEL + n];
      }
      out[(size_t)orow * ldo + n] = v;
    }
  }
}

// ---------------------------------------------------------------------------
// pack x_proj weights (24, 56, 768) -> zero-padded (24, 64, 768)
// ---------------------------------------------------------------------------
__global__ void pack_xproj_kernel(const float* __restrict__ w, float* __restrict__ wp) {
  int idx = blockIdx.x * blockDim.x + threadIdx.x;
  if (idx >= DEPTH * NXPAD * D_INNER) return;
  int k    = idx % D_INNER;
  int rest = idx / D_INNER;
  int n    = rest % NXPAD;
  int li   = rest / NXPAD;
  wp[idx] = (n < NXPROJ) ? w[((size_t)li * NXPROJ + n) * D_INNER + k] : 0.0f;
}

// ---------------------------------------------------------------------------
// im2col for 16x16 stride-16 patch embed: out[m, k], m=b*196+l, k=c*256+ph*16+pw
// ---------------------------------------------------------------------------
__global__ void im2col_kernel(const float* __restrict__ x, float* __restrict__ col) {
  int idx = blockIdx.x * blockDim.x + threadIdx.x;
  if (idx >= M_ROWS * 768) return;
  int k = idx % 768;
  int m = idx / 768;
  int b = m / L_TOK, l = m % L_TOK;
  int gh = l / 14, gw = l % 14;
  int c = k / 256, r = k % 256;
  int ph = r / 16, pw = r % 16;
  col[idx] = x[((b * 3 + c) * 224 + gh * 16 + ph) * 224 + (gw * 16 + pw)];
}

// ---------------------------------------------------------------------------
// residual update + RMSNorm + optional permute-gather store.  One wave per row.
// ---------------------------------------------------------------------------
__global__ void prenorm_kernel(const float* __restrict__ h, float* __restrict__ residual,
                               const float* __restrict__ nw, float* __restrict__ xn,
                               int first, int rotate) {
  int row = blockIdx.x;            // 0..783
  int lane = threadIdx.x;          // 32 lanes, 12 elems each
  float vals[12];
  float ss = 0.0f;
#pragma unroll
  for (int i = 0; i < 12; ++i) {
    int d = lane + i * 32;
    float r = first ? h[row * D_MODEL + d]
                    : residual[row * D_MODEL + d] + h[row * D_MODEL + d];
    residual[row * D_MODEL + d] = r;
    vals[i] = r;
    ss += r * r;
  }
#pragma unroll
  for (int off = 16; off > 0; off >>= 1) ss += __shfl_xor(ss, off, 32);
  float inv = rsqrtf(ss / (float)D_MODEL + EPSF);
  int b = row / L_TOK, l = row % L_TOK;
  int orow = rotate ? (b * L_TOK + rot_l(l)) : row;
#pragma unroll
  for (int i = 0; i < 12; ++i) {
    int d = lane + i * 32;
    xn[orow * D_MODEL + d] = vals[i] * inv * nw[d];
  }
}

// ---------------------------------------------------------------------------
// causal depthwise conv (width 4) + bias + SiLU.  Input = first half of xz.
// ---------------------------------------------------------------------------
__global__ void conv_silu_kernel(const float* __restrict__ xz,
                                 const float* __restrict__ cw,
                                 const float* __restrict__ cb,
                                 float* __restrict__ xin) {
  int idx = blockIdx.x * blockDim.x + threadIdx.x;
  if (idx >= M_ROWS * D_INNER) return;
  int e = idx % D_INNER;
  int m = idx / D_INNER;
  int l = m % L_TOK;
  float acc = cb[e];
#pragma unroll
  for (int t = 0; t < 4; ++t) {
    int ls = l - 3 + t;
    if (ls >= 0) acc += cw[e * 4 + t] * xz[(size_t)(m - 3 + t) * (2 * D_INNER) + e];
  }
  xin[idx] = acc / (1.0f + __expf(-acc));
}

// ---------------------------------------------------------------------------
// selective scan, one lane per (b, d, s).  The state recurrence
//   hs[s] = exp(dt*A[s])*hs[s] + dt*u*Bm[s]
// is independent across the 16 states; only y = sum_s hs[s]*Cm[s] couples
// them, done as a 4-step shfl_xor butterfly inside each 16-lane group
// (s = lane & 15, so xor masks 1/2/4/8 never cross the group).
// 49152 lanes total (16x the parallelism of one-thread-per-(b,d)), per-step
// critical path = 1 exp + 2 fma + 4 shfl-adds.  Next step's operand loads are
// issued before the current reduction so memory latency hides behind the
// recurrence.
// ---------------------------------------------------------------------------
__global__ void scan_kernel(const float* __restrict__ xin,
                            const float* __restrict__ dtbuf,
                            const float* __restrict__ xdbl,   // ld = NXPAD
                            const float* __restrict__ A_log,
                            const float* __restrict__ Dp,
                            float* __restrict__ ybuf) {
  int t  = blockIdx.x * blockDim.x + threadIdx.x;   // (b, d, s)
  int s  = t & (D_STATE - 1);
  int bd = t >> 4;                                  // b*D_INNER + d
  int b  = bd / D_INNER;
  int d  = bd % D_INNER;

  float Av = -__expf(A_log[d * D_STATE + s]);
  float Dv = Dp[d];
  float hs = 0.0f;

  const float* __restrict__ urow  = xin   + (size_t)(b * L_TOK) * D_INNER + d;
  const float* __restrict__ dtrow = dtbuf + (size_t)(b * L_TOK) * D_INNER + d;
  const float* __restrict__ bcrow = xdbl  + (size_t)(b * L_TOK) * NXPAD + DT_RANK;
  float* __restrict__       yrow  = ybuf  + (size_t)(b * L_TOK) * D_INNER + d;

  // prologue loads for step 0
  float u  = urow[0];
  float dt = dtrow[0];
  float bm = bcrow[s];
  float cm = bcrow[D_STATE + s];

  for (int l = 0; l < L_TOK; ++l) {
    // software pipeline: issue step l+1 loads before the step-l reduction
    float un = 0.0f, dtn = 0.0f, bmn = 0.0f, cmn = 0.0f;
    if (l + 1 < L_TOK) {
      un  = urow[(size_t)(l + 1) * D_INNER];
      dtn = dtrow[(size_t)(l + 1) * D_INNER];
      bmn = bcrow[(size_t)(l + 1) * NXPAD + s];
      cmn = bcrow[(size_t)(l + 1) * NXPAD + D_STATE + s];
    }
    float dA = __expf(dt * Av);
    hs = dA * hs + (dt * u) * bm;
    float p = hs * cm;
    p += __shfl_xor(p, 1, 32);
    p += __shfl_xor(p, 2, 32);
    p += __shfl_xor(p, 4, 32);
    p += __shfl_xor(p, 8, 32);
    if (s == 0) yrow[(size_t)l * D_INNER] = p + u * Dv;
    u = un; dt = dtn; bm = bmn; cm = cmn;
  }
}

// ---------------------------------------------------------------------------
// RMSNorm(y) * ssm_norm_w * SiLU(z).  One wave per row; safe in-place on y.
// ---------------------------------------------------------------------------
__global__ void gate_norm_kernel(const float* __restrict__ y,
                                 const float* __restrict__ xz,
                                 const float* __restrict__ snw,
                                 float* __restrict__ yact) {
  int row = blockIdx.x;
  int lane = threadIdx.x;
  float vals[24];
  float ss = 0.0f;
#pragma unroll
  for (int i = 0; i < 24; ++i) {
    int e = lane + i * 32;
    float v = y[row * D_INNER + e];
    vals[i] = v;
    ss += v * v;
  }
#pragma unroll
  for (int off = 16; off > 0; off >>= 1) ss += __shfl_xor(ss, off, 32);
  float inv = rsqrtf(ss / (float)D_INNER + EPSF);
#pragma unroll
  for (int i = 0; i < 24; ++i) {
    int e = lane + i * 32;
    float z = xz[row * (2 * D_INNER) + D_INNER + e];
    float sz = z / (1.0f + __expf(-z));
    yact[row * D_INNER + e] = vals[i] * inv * snw[e] * sz;
  }
}

// ---------------------------------------------------------------------------
// mean-pool over L, then classifier head
// ---------------------------------------------------------------------------
__global__ void pool_kernel(const float* __restrict__ xn, float* __restrict__ pooled) {
  int t = blockIdx.x * blockDim.x + threadIdx.x;
  if (t >= BATCH * D_MODEL) return;
  int b = t / D_MODEL, d = t % D_MODEL;
  float s = 0.0f;
  for (int l = 0; l < L_TOK; ++l) s += xn[(b * L_TOK + l) * D_MODEL + d];
  pooled[t] = s / (float)L_TOK;
}

__global__ void head_kernel(const float* __restrict__ pooled,
                            const float* __restrict__ hw,
                            const float* __restrict__ hb,
                            float* __restrict__ out) {
  int t = blockIdx.x * blockDim.x + threadIdx.x;
  if (t >= BATCH * 1000) return;
  int b = t / 1000, c = t % 1000;
  float s = hb[c];
  for (int d = 0; d < D_MODEL; ++d) s += pooled[b * D_MODEL + d] * hw[c * D_MODEL + d];
  out[t] = s;
}

// ---------------------------------------------------------------------------
// host sequencing
// ---------------------------------------------------------------------------
static inline void launch_gemm(const float* A, int lda, const float* W,
                               float* out, int ldo, int M, int N, int K,
                               int mode, const float* aux0, const float* aux1,
                               hipStream_t stream) {
  dim3 grid(M / 16, N / 64);
  gemm_wmma_f32<<<grid, 32, 0, stream>>>(A, lda, W, out, ldo, K, mode, aux0, aux1);
}

extern "C" void kernel_launch(void* const* d_in, const int* in_sizes, int n_in,
                              void* d_out, int out_size, void* d_ws, size_t ws_size,
                              hipStream_t stream) {
  const float* x          = (const float*)d_in[0];
  /* d_in[1] = ids_keep (unused by reference) */
  const float* patch_w    = (const float*)d_in[2];   // (384, 768)
  const float* patch_b    = (const float*)d_in[3];   // (384,)
  const float* pos_embed  = (const float*)d_in[4];   // (1, 196, 384)
  const float* norm_w     = (const float*)d_in[5];   // (24, 384)
  const float* in_proj_w  = (const float*)d_in[6];   // (24, 1536, 384)
  const float* conv1d_w   = (const float*)d_in[7];   // (24, 768, 4)
  const float* conv1d_b   = (const float*)d_in[8];   // (24, 768)
  const float* x_proj_w   = (const float*)d_in[9];   // (24, 56, 768)
  const float* dt_proj_w  = (const float*)d_in[10];  // (24, 768, 24)
  const float* dt_proj_b  = (const float*)d_in[11];  // (24, 768)
  const float* A_log      = (const float*)d_in[12];  // (24, 768, 16)
  const float* Dparam     = (const float*)d_in[13];  // (24, 768)
  const float* ssm_norm_w = (const float*)d_in[14];  // (24, 768)
  const float* out_proj_w = (const float*)d_in[15];  // (24, 384, 768)
  const float* norm_f_w   = (const float*)d_in[16];  // (384,)
  const float* head_w     = (const float*)d_in[17];  // (1000, 384)
  const float* head_b     = (const float*)d_in[18];  // (1000,)
  float* out = (float*)d_out;

  // workspace layout (floats)
  float* ws       = (float*)d_ws;
  float* h        = ws;  ws += M_ROWS * D_MODEL;          // 301056
  float* residual = ws;  ws += M_ROWS * D_MODEL;          // 301056
  float* xn       = ws;  ws += M_ROWS * D_MODEL;          // 301056
  float* xz       = ws;  ws += M_ROWS * 2 * D_INNER;      // 1204224
  float* xin      = ws;  ws += M_ROWS * D_INNER;          // 602112
  float* xdbl     = ws;  ws += M_ROWS * NXPAD;            // 50176
  float* dtb      = ws;  ws += M_ROWS * D_INNER;          // 602112
  float* ybuf     = ws;  ws += M_ROWS * D_INNER;          // 602112
  float* pooled   = ws;  ws += BATCH * D_MODEL;           // 1536
  float* xprojpad = ws;  ws += DEPTH * NXPAD * D_INNER;   // 1179648  (~19.6 MiB total)
  float* im2col   = xz;  // aliases xz: used only before the layer loop

  // one-time (per call) weight pad for x_proj: N 56 -> 64
  {
    int n = DEPTH * NXPAD * D_INNER;
    pack_xproj_kernel<<<(n + 255) / 256, 256, 0, stream>>>(x_proj_w, xprojpad);
  }

  // patch embed: im2col + GEMM with fused bias + pos_embed
  {
    int n = M_ROWS * 768;
    im2col_kernel<<<(n + 255) / 256, 256, 0, stream>>>(x, im2col);
    launch_gemm(im2col, 768, patch_w, h, D_MODEL, M_ROWS, D_MODEL, 768,
                3, patch_b, pos_embed, stream);
  }

  for (int li = 0; li < DEPTH; ++li) {
    int rotate = li & 1;
    prenorm_kernel<<<M_ROWS, 32, 0, stream>>>(h, residual, norm_w + li * D_MODEL, xn,
                                              li == 0, rotate);
    // in_proj: (784,384) x (384,1536)
    launch_gemm(xn, D_MODEL, in_proj_w + (size_t)li * 2 * D_INNER * D_MODEL,
                xz, 2 * D_INNER, M_ROWS, 2 * D_INNER, D_MODEL, 0, nullptr, nullptr, stream);
    // depthwise causal conv + SiLU
    {
      int n = M_ROWS * D_INNER;
      conv_silu_kernel<<<(n + 255) / 256, 256, 0, stream>>>(
          xz, conv1d_w + (size_t)li * D_INNER * 4, conv1d_b + (size_t)li * D_INNER, xin);
    }
    // x_proj: (784,768) x (768,64-padded) -> xdbl (ld 64)
    launch_gemm(xin, D_INNER, xprojpad + (size_t)li * NXPAD * D_INNER,
                xdbl, NXPAD, M_ROWS, NXPAD, D_INNER, 0, nullptr, nullptr, stream);
    // dt_proj: (784,24) x (24,768), fused +bias, softplus  (A ld = 64, K = 24)
    launch_gemm(xdbl, NXPAD, dt_proj_w + (size_t)li * D_INNER * DT_RANK,
                dtb, D_INNER, M_ROWS, D_INNER, DT_RANK,
                1, dt_proj_b + (size_t)li * D_INNER, nullptr, stream);
    // selective scan: one lane per (b, d, s) = 49152 lanes, 192 blocks
    scan_kernel<<<(BATCH * D_INNER * D_STATE) / 256, 256, 0, stream>>>(
        xin, dtb, xdbl, A_log + (size_t)li * D_INNER * D_STATE,
        Dparam + (size_t)li * D_INNER, ybuf);
    // RMSNorm * w * SiLU(z)  (in place on ybuf)
    gate_norm_kernel<<<M_ROWS, 32, 0, stream>>>(ybuf, xz,
                                                ssm_norm_w + (size_t)li * D_INNER, ybuf);
    // out_proj: (784,768) x (768,384), un-permute fused into store for rotate layers
    launch_gemm(ybuf, D_INNER, out_proj_w + (size_t)li * D_MODEL * D_INNER,
                h, D_MODEL, M_ROWS, D_MODEL, D_INNER,
                rotate ? 2 : 0, nullptr, nullptr, stream);
  }

  // final residual + norm, pool, head
  prenorm_kernel<<<M_ROWS, 32, 0, stream>>>(h, residual, norm_f_w, xn, 0, 0);
  pool_kernel<<<(BATCH * D_MODEL + 255) / 256, 256, 0, stream>>>(xn, pooled);
  head_kernel<<<(BATCH * 1000 + 255) / 256, 256, 0, stream>>>(pooled, head_w, head_b, out);
}